// GaussianDerivativeESPLayer_90159953478162
// MI455X (gfx1250) — compile-verified
//
#include <hip/hip_runtime.h>
#include <math.h>

typedef __attribute__((ext_vector_type(2))) float v2f;
typedef __attribute__((ext_vector_type(8))) float v8f;
typedef __attribute__((ext_vector_type(4))) unsigned int u32x4;
typedef __attribute__((ext_vector_type(4))) int i32x4;
typedef __attribute__((ext_vector_type(8))) int i32x8;

#define IH 192
#define IW 192
#define IC 64
#define OB 4
#define OH 186
#define OW 186
#define OC 12
#define KI 7          // integrator 7x7
#define TPX 6         // output pixels per block (186 = 6*31)
#define TCOL (TPX + KI - 1)   // 12 tile columns
#define CCH 16        // channels per TDM chunk (4 chunks of 16)

// ---------------------------------------------------------------------------
// Stage 1: 6 depthwise Gaussian-derivative convs (SAME, zero pad) for one
// (sigma, batch). Weights identical across channels -> staged in LDS.
// Output L[n][y][x][c], pre-scaled by sqrt(C(j,k)).
// ---------------------------------------------------------------------------
__global__ __launch_bounds__(256) void dwconv6_kernel(
    const float* __restrict__ in,     // [IH,IW,IC] (batch-offset applied)
    const float* __restrict__ kern,   // [6,kk,kk,IC,1]; identical across c
    float* __restrict__ L,            // [6,IH,IW,IC]
    int r)
{
    __shared__ float sk[6 * 13 * 13];
    const int kk  = 2 * r + 1;
    const int kk2 = kk * kk;
    const int nw  = 6 * kk2;
    for (int i = threadIdx.x; i < nw; i += blockDim.x)
        sk[i] = kern[(size_t)i * IC];   // c=0 slice (tiled across channels)
    __syncthreads();

    const int g = blockIdx.x * blockDim.x + threadIdx.x;   // < IH*IW*IC
    const int c = g & (IC - 1);
    const int x = (g >> 6) % IW;
    const int y = g / (IW * IC);

    float acc[6] = {0.f, 0.f, 0.f, 0.f, 0.f, 0.f};
    for (int a = 0; a < kk; ++a) {
        const int yy = y + a - r;
        if (yy < 0 || yy >= IH) continue;
        const float* row  = in + ((size_t)yy * IW) * IC + c;
        const float* krow = &sk[a * kk];
        for (int b = 0; b < kk; ++b) {
            const int xx = x + b - r;
            if (xx < 0 || xx >= IW) continue;
            const float v = row[(size_t)xx * IC];
#pragma unroll
            for (int n = 0; n < 6; ++n)
                acc[n] += v * krow[n * kk2 + b];
        }
    }
    // components: [ (j0,k0), (j1,k0), (j1,k1), (j2,k0), (j2,k1), (j2,k2) ]
    const float sc[6] = {1.f, 1.f, 1.f, 1.f, 1.41421356237309515f, 1.f};
#pragma unroll
    for (int n = 0; n < 6; ++n)
        L[(((size_t)n * IH + y) * IW + x) * IC + c] = acc[n] * sc[n];
}

// ---------------------------------------------------------------------------
// Stage 2: TDM-staged Gram integration.
//   TENSOR_LOAD_TO_LDS stages a [6 comp][7 row][12 col][16 ch] 4D tile.
//   Each of 3 waves computes the 16x16 fp32 WMMA Gram for 2 output pixels
//   (rows/cols 0..5 = pixel0, 8..13 = pixel1; A and B register images of
//   W / W^T coincide in the CDNA5 f32 layout). Newton-identity ESP epilogue.
//
// LDS tile packing (done by TDM): linear index
//   idx = ((comp*7 + u)*12 + col)*16 + c,  bytes = 4*idx + 8*(idx>>6)
// (pad_enable: 2 dwords after every 64 dwords -> keeps 8B alignment and
//  staggers the comp stride off the 64-bank period).
// ---------------------------------------------------------------------------
#define TILE_DWORDS (6 * KI * TCOL * CCH)              // 8064
#define TILE_FLOATS (TILE_DWORDS + 2 * (TILE_DWORDS / 64))

__global__ __launch_bounds__(96) void integrate_esp_kernel(
    const float* __restrict__ L,      // [6,IH,IW,IC]
    const float* __restrict__ dgint,  // [7,7,IC,1]
    float* __restrict__ out,          // [OB,OH,OW,OC]
    int b, int chbase)
{
    __shared__ float tile[TILE_FLOATS];
    __shared__ float sdg[KI * KI];
    __shared__ float G[3][2][36];

    const int tid  = threadIdx.x;
    const int wv   = tid >> 5;        // wave 0..2 -> pixel pair
    const int lane = tid & 31;

    for (int i = tid; i < KI * KI; i += blockDim.x)
        sdg[i] = sqrtf(dgint[(size_t)i * IC]);   // integrator is positive
    __syncthreads();

    const int p  = blockIdx.y;
    const int qb = blockIdx.x * TPX;

    const int m = lane & 15;
    const int h = lane >> 4;          // K-half: 0 -> k0/k1, 1 -> k2/k3
    const bool valid = (m < 6) || (m >= 8 && m < 14);
    const float lanemask = valid ? 1.0f : 0.0f;           // branch-free
    const int comp    = valid ? ((m < 6) ? m : (m - 8)) : 0;
    const int colbase = 2 * wv + ((m < 8) ? 0 : 1);       // tile column base

    // ---- Tensor DMA descriptor (constant parts) -------------------------
    const unsigned lds_base = (unsigned)(uintptr_t)(void*)&tile[0];
    u32x4 g0; i32x8 g1; i32x4 g2; i32x4 g3; i32x8 g4;
    // group1: data_size=4B(2), pad_enable, pad_interval=64dw(5), pad=2dw(1)
    g1[0] = (2 << 16) | (1 << 20) | (5 << 22) | (1 << 25);
    g1[1] = (CCH & 0xFFFF) << 16;                 // tensor_dim0 = 16
    g1[2] = (TCOL & 0xFFFF) << 16;                // tensor_dim1 = 12
    g1[3] = (CCH << 16);                          // tile_dim0 = 16
    g1[4] = TCOL | (KI << 16);                    // tile_dim1=12, tile_dim2=7
    g1[5] = IC;                                   // dim0 stride = 64 (col)
    g1[6] = ((IW * IC) & 0xFFFF) << 16;           // dim1 stride lo16 (row)
    g1[7] = (IW * IC) >> 16;                      // dim1 stride hi
    g2[0] = KI;                                   // tensor_dim2 = 7
    g2[1] = 6;                                    // tensor_dim3 = 6
    g2[2] = IH * IW * IC;                         // dim2 stride (comp)
    g2[3] = (6 << 16);                            // tile_dim3 = 6 comps
    g3[0] = 0; g3[1] = 0; g3[2] = 0; g3[3] = 0;
    g4[0] = 0; g4[1] = 0; g4[2] = 0; g4[3] = 0;   // zero-filled trailing group
    g4[4] = 0; g4[5] = 0; g4[6] = 0; g4[7] = 0;

    v8f acc = {};
    for (int ch = 0; ch < (IC / CCH); ++ch) {     // 4 channel chunks
        // ---- stage tile via Tensor Data Mover (wave 0 issues) ----------
        const unsigned long long ga =
            (unsigned long long)(uintptr_t)L +
            4ull * (((size_t)p * IW + qb) * IC + ch * CCH);
        g0[0] = 1u;                               // count=1, user mode
        g0[1] = lds_base;
        g0[2] = (unsigned)(ga & 0xFFFFFFFFull);
        g0[3] = (unsigned)((ga >> 32) & 0x01FFFFFFull) | 0x80000000u; // type=2
        if (tid < 32) {
            __builtin_amdgcn_tensor_load_to_lds(g0, g1, g2, g3, g4, 0);
            __builtin_amdgcn_s_wait_tensorcnt(0);
        }
        __syncthreads();                          // tile visible to all waves

        // ---- Gram accumulation over this chunk -------------------------
        for (int u = 0; u < KI; ++u) {
#pragma unroll
            for (int v = 0; v < KI; ++v) {
                const float scale = sdg[u * KI + v] * lanemask;
                const int ib = ((comp * KI + u) * TCOL + (colbase + v)) * CCH
                               + 2 * h;
#pragma unroll
                for (int t = 0; t < CCH / 4; ++t) {   // 4 WMMAs (K=4 each)
                    const int idx = ib + 4 * t;
                    const float2 d = *(const float2*)(
                        (const char*)tile + ((idx << 2) + ((idx >> 6) << 3)));
                    v2f a;
                    a.x = d.x * scale;
                    a.y = d.y * scale;
                    acc = __builtin_amdgcn_wmma_f32_16x16x4_f32(
                        false, a, false, a, (short)0, acc, false, false);
                }
            }
        }
        __syncthreads();                          // protect tile re-fill
    }

    // ---- scatter Gram entries: acc[g] holds D[g + 8*h][lane&15] ---------
    const int n = lane & 15;
    if (h == 0 && n < 6) {
#pragma unroll
        for (int g = 0; g < 6; ++g) G[wv][0][g * 6 + n] = acc[g];
    }
    if (h == 1 && n >= 8 && n < 14) {
#pragma unroll
        for (int g = 0; g < 6; ++g) G[wv][1][g * 6 + (n - 8)] = acc[g];
    }
    __syncthreads();

    // ---- Newton-identity ESPs: one thread per output pixel --------------
    if (tid < TPX) {
        const float* Gm = G[tid >> 1][tid & 1];
        const float EPS = 2.220446049250313e-16f;
        float o[6];

        o[0] = fabsf(Gm[0]) + EPS;                        // j=0

        {   // j=1 : 2x2 from components {1,2}
            const float a11 = Gm[1 * 6 + 1], a12 = Gm[1 * 6 + 2];
            const float a21 = Gm[2 * 6 + 1], a22 = Gm[2 * 6 + 2];
            const float p1 = a11 + a22;
            const float p2 = a11 * a11 + 2.f * a12 * a21 + a22 * a22;
            const float e1 = p1;
            const float e2 = 0.5f * (p1 * e1 - p2);
            o[1] = 10.0f * (fabsf(e1) + EPS);
            o[2] = 10.0f * sqrtf(fabsf(e2) + EPS);
        }
        {   // j=2 : 3x3 from components {3,4,5}
            float A[3][3];
#pragma unroll
            for (int i = 0; i < 3; ++i)
#pragma unroll
                for (int jj = 0; jj < 3; ++jj)
                    A[i][jj] = Gm[(3 + i) * 6 + (3 + jj)];
            float p1 = A[0][0] + A[1][1] + A[2][2];
            float p2 = 0.f, p3 = 0.f;
#pragma unroll
            for (int i = 0; i < 3; ++i)
#pragma unroll
                for (int k = 0; k < 3; ++k) {
                    p2 += A[i][k] * A[k][i];
#pragma unroll
                    for (int jj = 0; jj < 3; ++jj)
                        p3 += A[i][k] * A[k][jj] * A[jj][i];
                }
            const float e1 = p1;
            const float e2 = 0.5f * (p1 * e1 - p2);
            const float e3 = (p1 * e2 - p2 * e1 + p3) * (1.0f / 3.0f);
            o[3] = 100.0f * (fabsf(e1) + EPS);
            o[4] = 100.0f * sqrtf(fabsf(e2) + EPS);
            o[5] = 100.0f * cbrtf(fabsf(e3) + EPS);
        }

        float* dst = out + (((size_t)b * OH + p) * OW + (qb + tid)) * OC + chbase;
#pragma unroll
        for (int i = 0; i < 6; ++i) dst[i] = o[i];
    }
}

// ---------------------------------------------------------------------------
// Host launcher: per (sigma, batch) pipeline, workspace reused sequentially.
// ws needs 6*192*192*64*4 = ~57 MB.
// ---------------------------------------------------------------------------
extern "C" void kernel_launch(void* const* d_in, const int* in_sizes, int n_in,
                              void* d_out, int out_size, void* d_ws, size_t ws_size,
                              hipStream_t stream) {
    const float* x   = (const float*)d_in[0];   // [4,192,192,64]
    const float* k0  = (const float*)d_in[1];   // [6,7,7,64,1]
    const float* k1  = (const float*)d_in[2];   // [6,13,13,64,1]
    const float* dgi = (const float*)d_in[3];   // [7,7,64,1]
    float* out = (float*)d_out;
    float* ws  = (float*)d_ws;

    const int total = IH * IW * IC;
    const dim3 g1((total + 255) / 256), b1(256);
    const dim3 g2(OW / TPX, OH), b2(96);        // 31 x 186 blocks, 3 waves

    for (int s = 0; s < 2; ++s) {
        const float* kern = (s == 0) ? k0 : k1;
        const int r = (s == 0) ? 3 : 6;
        for (int b = 0; b < OB; ++b) {
            dwconv6_kernel<<<g1, b1, 0, stream>>>(
                x + (size_t)b * IH * IW * IC, kern, ws, r);
            integrate_esp_kernel<<<g2, b2, 0, stream>>>(
                ws, dgi, out, b, s * 6);
        }
    }
}